// Attention_60344290509655
// MI455X (gfx1250) — compile-verified
//
#include <hip/hip_runtime.h>

// Problem constants (from reference): B=4, T=512, INPUT_DIM=128, HIDDEN=64, DEMO=12
constexpr int B = 4, T = 512, D = 128, H = 64, DEMO = 12;

typedef float v2f __attribute__((ext_vector_type(2)));
typedef float v8f __attribute__((ext_vector_type(8)));

__device__ __forceinline__ float tanh_fast(float x) {
#if __has_builtin(__builtin_amdgcn_tanhf)
  // gfx1250 hardware v_tanh_f32 (TRANS pipe, co-executes with VALU FMA)
  return __builtin_amdgcn_tanhf(x);
#else
  // fallback: tanh(x) = 1 - 2/(exp(2x)+1) via v_exp_f32
  float e2 = __expf(2.0f * x);
  return 1.0f - 2.0f / (e2 + 1.0f);
#endif
}

// ---------------------------------------------------------------------------
// K1: qc[b,t,h] = input[b,t,:]@Wt[:,h] + demo[b,:]@Wd[:,h] + bh[h]
//     kb[b,t,h] = input[b,t,:]@Wx[:,h]
// ---------------------------------------------------------------------------
__global__ void qk_kernel(const float* __restrict__ input, const float* __restrict__ demo,
                          const float* __restrict__ Wt, const float* __restrict__ Wx,
                          const float* __restrict__ Wd, const float* __restrict__ bh,
                          float* __restrict__ qc, float* __restrict__ kb) {
  int idx = blockIdx.x * blockDim.x + threadIdx.x;   // [0, B*T*H)
  int h = idx & (H - 1);
  int t = (idx >> 6) & (T - 1);
  int b = idx >> 15;
  float aq = bh[h];
  #pragma unroll
  for (int j = 0; j < DEMO; ++j) aq = fmaf(demo[b * DEMO + j], Wd[j * H + h], aq);
  float ak = 0.0f;
  const float* xr = input + (size_t)(b * T + t) * D;  // uniform per wave -> scalar loads
  #pragma unroll 4
  for (int j = 0; j < D; ++j) {
    float x = xr[j];
    aq = fmaf(x, Wt[j * H + h], aq);
    ak = fmaf(x, Wx[j * H + h], ak);
  }
  qc[idx] = aq;
  kb[idx] = ak;
}

// ---------------------------------------------------------------------------
// K2: raw score e[b,t,s] = ba + sum_h Wa[h] * tanh(qc[b,t,h] + kb[b,s,h])
// 16x16 (t,s) tile per 256-thread block; q/k tiles staged in LDS.
// This is the dominant cost: 4*512*512*64 = 67M tanh-FMA ops.
// ---------------------------------------------------------------------------
__global__ void score_kernel(const float* __restrict__ qc, const float* __restrict__ kb,
                             const float* __restrict__ Wa, const float* __restrict__ ba,
                             float* __restrict__ e) {
  __shared__ float qs[16][H + 1];   // +1 pad -> conflict-free column reads
  __shared__ float ks[16][H + 1];
  __shared__ float was[H];
  int s0 = blockIdx.x * 16, t0 = blockIdx.y * 16, b = blockIdx.z;
  int tid = threadIdx.x;
  for (int i = tid; i < 16 * H; i += 256) {
    int r = i >> 6, c = i & (H - 1);
    qs[r][c] = qc[(size_t)(b * T + t0 + r) * H + c];
    ks[r][c] = kb[(size_t)(b * T + s0 + r) * H + c];
  }
  if (tid < H) was[tid] = Wa[tid];
  __syncthreads();
  int tt = tid >> 4, ss = tid & 15;
  float acc = ba[0];
  #pragma unroll
  for (int h = 0; h < H; ++h)
    acc = fmaf(was[h], tanh_fast(qs[tt][h] + ks[ss][h]), acc);
  e[(size_t)b * T * T + (size_t)(t0 + tt) * T + (s0 + ss)] = acc;
}

// ---------------------------------------------------------------------------
// K3: per-row: m = max over FULL row, v = exp(x-m), causal mask (s<=t),
//     normalize by (sum + 1e-7). In place on the e output region.
// ---------------------------------------------------------------------------
__global__ void softmax_kernel(float* __restrict__ e) {
  int t = blockIdx.x, b = blockIdx.y, tid = threadIdx.x;
  float* er = e + ((size_t)b * T + t) * T;
  __shared__ float red[256];
  float x0 = er[tid], x1 = er[tid + 256];
  red[tid] = fmaxf(x0, x1);
  __syncthreads();
  for (int off = 128; off > 0; off >>= 1) {
    if (tid < off) red[tid] = fmaxf(red[tid], red[tid + off]);
    __syncthreads();
  }
  float m = red[0];
  __syncthreads();
  float v0 = (tid       <= t) ? __expf(x0 - m) : 0.0f;  // mask AFTER exp (matches ref)
  float v1 = (tid + 256 <= t) ? __expf(x1 - m) : 0.0f;
  red[tid] = v0 + v1;
  __syncthreads();
  for (int off = 128; off > 0; off >>= 1) {
    if (tid < off) red[tid] += red[tid + off];
    __syncthreads();
  }
  float inv = 1.0f / (red[0] + 1e-7f);
  er[tid]       = v0 * inv;
  er[tid + 256] = v1 * inv;
}

// ---------------------------------------------------------------------------
// K4: v[b] = e[b] @ input[b]  ([512x512]x[512x128]) via V_WMMA_F32_16X16X4_F32.
// One wave per 16x16 output tile; K-loop of 128 WMMAs.
// f32 A layout (16x4): vgpr0 = K=0 (lanes 0-15) / K=2 (lanes 16-31), vgpr1 = K=1/K=3.
// f32 C/D layout: vgpr i holds M=i (lanes 0-15) / M=i+8 (lanes 16-31), N=lane&15.
// ---------------------------------------------------------------------------
__global__ void __launch_bounds__(32) av_wmma_kernel(const float* __restrict__ e,
                                                     const float* __restrict__ input,
                                                     float* __restrict__ vout) {
  int n0 = blockIdx.x * 16, t0 = blockIdx.y * 16, b = blockIdx.z;
  int lane = threadIdx.x;
  int col = lane & 15;     // M-row for A, N-col for B/D
  int hi  = lane >> 4;     // K/M half select
  const float* E = e     + (size_t)b * T * T;
  const float* X = input + (size_t)b * T * D;
  v8f c = {};
  for (int kk = 0; kk < T; kk += 4) {
    int ka = kk + 2 * hi;
    v2f a, bb;
    a.x  = E[(size_t)(t0 + col) * T + ka];       // A[M=col][K=ka]
    a.y  = E[(size_t)(t0 + col) * T + ka + 1];   // A[M=col][K=ka+1]
    bb.x = X[(size_t)ka * D + n0 + col];         // B[K=ka][N=col]
    bb.y = X[(size_t)(ka + 1) * D + n0 + col];   // B[K=ka+1][N=col]
    c = __builtin_amdgcn_wmma_f32_16x16x4_f32(false, a, false, bb, (short)0, c,
                                              false, false);
  }
  #pragma unroll
  for (int i = 0; i < 8; ++i)
    vout[(size_t)b * T * D + (size_t)(t0 + i + 8 * hi) * D + n0 + col] = c[i];
}

// ---------------------------------------------------------------------------
extern "C" void kernel_launch(void* const* d_in, const int* in_sizes, int n_in,
                              void* d_out, int out_size, void* d_ws, size_t ws_size,
                              hipStream_t stream) {
  const float* input = (const float*)d_in[0];
  const float* demo  = (const float*)d_in[1];
  const float* Wt    = (const float*)d_in[2];
  const float* Wx    = (const float*)d_in[3];
  const float* Wd    = (const float*)d_in[4];
  const float* bh    = (const float*)d_in[5];
  const float* Wa    = (const float*)d_in[6];
  const float* ba    = (const float*)d_in[7];

  float* vout = (float*)d_out;                   // [B,T,D] = 262144 floats
  float* eout = vout + (size_t)B * T * D;        // [B,T,T] = 1048576 floats

  float* qc = (float*)d_ws;                      // [B,T,H]
  float* kb = qc + (size_t)B * T * H;            // [B,T,H]  (1 MB total scratch)

  qk_kernel<<<(B * T * H) / 256, 256, 0, stream>>>(input, demo, Wt, Wx, Wd, bh, qc, kb);

  dim3 g2(T / 16, T / 16, B);                    // 4096 blocks
  score_kernel<<<g2, 256, 0, stream>>>(qc, kb, Wa, ba, eout);

  dim3 g3(T, B);                                 // 2048 rows
  softmax_kernel<<<g3, 256, 0, stream>>>(eout);

  dim3 g4(D / 16, T / 16, B);                    // 1024 tiles, 1 wave each
  av_wmma_kernel<<<g4, 32, 0, stream>>>(eout, input, vout);
}